// CenterLossLayer_39436389712535
// MI455X (gfx1250) — compile-verified
//
#include <hip/hip_runtime.h>

#define ALPHA_CENTER 0.5f

typedef __attribute__((ext_vector_type(2))) float v2f;
typedef __attribute__((ext_vector_type(4))) float v4f;
typedef __attribute__((ext_vector_type(8))) float v8f;

// ---------------------------------------------------------------------------
// Kernel 1: zero per-class counts in workspace.
// ---------------------------------------------------------------------------
__global__ __launch_bounds__(256) void zero_counts_kernel(int* __restrict__ counts, int C) {
  int i = blockIdx.x * 256 + threadIdx.x;
  if (i < C) counts[i] = 0;
}

// ---------------------------------------------------------------------------
// Kernel 2: stream the dense one-hot (1.6 GB -> dominant, pure HBM bandwidth).
// Non-temporal B128 loads so the one-hot stream does not evict the 192 MB L2
// working set (centers / new_centers) that the atomic scatter wants resident.
// Exactly one element per row is 1.0, so a plain store of the label is safe.
// ---------------------------------------------------------------------------
__global__ __launch_bounds__(256) void extract_labels_kernel(
    const float* __restrict__ onehot, int* __restrict__ labels,
    int* __restrict__ counts, int C4) {
  int row  = blockIdx.y;
  int col4 = blockIdx.x * 256 + threadIdx.x;
  if (col4 >= C4) return;
  const v4f* p = (const v4f*)onehot + (size_t)row * C4 + col4;
  v4f v = __builtin_nontemporal_load(p);
  int hit = -1;
  if (v.x > 0.5f) hit = 0;
  if (v.y > 0.5f) hit = 1;
  if (v.z > 0.5f) hit = 2;
  if (v.w > 0.5f) hit = 3;
  if (hit >= 0) {
    int col = col4 * 4 + hit;
    labels[row] = col;
    atomicAdd(&counts[col], 1);
  }
}

// ---------------------------------------------------------------------------
// Kernel 3: new_centers = centers (classes not present in the batch are
// unchanged; present classes get atomically updated by kernel 4).
// ---------------------------------------------------------------------------
__global__ __launch_bounds__(256) void copy_centers_kernel(
    const float* __restrict__ src, float* __restrict__ dst, size_t n4) {
  size_t i = (size_t)blockIdx.x * 256 + threadIdx.x;
  if (i < n4) ((v4f*)dst)[i] = ((const v4f*)src)[i];
}

// ---------------------------------------------------------------------------
// Kernel 4: one wave32 per 16 samples.
//  - Residual R[16][128] = x - centers[label].
//  - acc(16x16,f32) += A(16x4) x A^T(4x16) over 32 K-chunks via
//    V_WMMA_F32_16X16X4_F32; diag(acc) = per-sample squared distance.
//    A layout (32-bit 16x4): lanes 0-15 hold M=0..15 with K={0,1} in the two
//    VGPRs, lanes 16-31 hold K={2,3}. For B=A^T (4x16) the per-lane register
//    contents are identical (symmetric Gram), so we pass `a` twice.
//  - Scatter ALPHA*(x-c)/(count+1) into new_centers[label] with f32 atomics
//    (per-sample division by the class count is linear in the sum).
// ---------------------------------------------------------------------------
__global__ __launch_bounds__(32) void per_sample_kernel(
    const float* __restrict__ x, const float* __restrict__ centers,
    const int* __restrict__ labels, const int* __restrict__ counts,
    float* __restrict__ result, float* __restrict__ new_centers) {
  const int D = 128;
  int lane  = threadIdx.x;       // 0..31, wave32
  int m     = lane & 15;         // matrix row this lane carries
  int khalf = lane >> 4;         // K-half (0 -> K{0,1}, 1 -> K{2,3})
  int s     = blockIdx.x * 16 + m;
  int lbl   = labels[s];
  const float* xr = x       + (size_t)s   * D;
  const float* cr = centers + (size_t)lbl * D;

  v8f acc = {};
#pragma unroll
  for (int chunk = 0; chunk < 32; ++chunk) {
    int k0 = chunk * 4 + khalf * 2;               // even -> 8B aligned
    v2f xa = *(const v2f*)(xr + k0);
    v2f ca = *(const v2f*)(cr + k0);
    v2f a  = xa - ca;
    acc = __builtin_amdgcn_wmma_f32_16x16x4_f32(
        /*neg_a=*/false, a, /*neg_b=*/false, a,
        /*c_mod=*/(short)0, acc, /*reuse_a=*/false, /*reuse_b=*/false);
  }

  // Diagonal extraction. C/D layout (16x16 f32, 8 VGPRs):
  //   VGPR r: lanes 0-15 -> (M=r,   N=lane), lanes 16-31 -> (M=r+8, N=lane-16)
  // So diag(i,i): i<8 at VGPR i lane i; i>=8 at VGPR i-8 lane i+16.
  int r = (lane < 8) ? lane : (lane - 24);        // lanes 8..23 -> no match
  float dval = 0.0f;
#pragma unroll
  for (int j = 0; j < 8; ++j)
    if (r == j) dval = acc[j];
  if (lane < 8)        result[blockIdx.x * 16 + lane]        = dval;
  else if (lane >= 24) result[blockIdx.x * 16 + (lane - 16)] = dval;

  // Scatter the centers update: each lane covers one sample, half of D.
  float invc = ALPHA_CENTER / ((float)counts[lbl] + 1.0f);
  float* nc  = new_centers + (size_t)lbl * D;
  int dbase  = khalf * 64;
#pragma unroll 4
  for (int dd = 0; dd < 64; ++dd) {
    int d2 = dbase + dd;
    atomicAdd(&nc[d2], invc * (xr[d2] - cr[d2]));  // -ALPHA*(c-x)/cnt
  }
}

// ---------------------------------------------------------------------------
extern "C" void kernel_launch(void* const* d_in, const int* in_sizes, int n_in,
                              void* d_out, int out_size, void* d_ws, size_t ws_size,
                              hipStream_t stream) {
  (void)n_in; (void)out_size; (void)ws_size;
  const float* x       = (const float*)d_in[0];   // [B, D]
  const float* onehot  = (const float*)d_in[1];   // [B, C]
  const float* centers = (const float*)d_in[2];   // [C, D]

  const int D = 128;
  const int B = in_sizes[0] / D;                  // 4096
  const int C = in_sizes[2] / D;                  // 100000

  float* result      = (float*)d_out;             // [B]
  float* new_centers = (float*)d_out + B;         // [C, D]

  int* counts = (int*)d_ws;                       // [C]
  int* labels = counts + C;                       // [B]

  // 1) zero counts
  zero_counts_kernel<<<(C + 255) / 256, 256, 0, stream>>>(counts, C);

  // 2) label + count extraction from dense one-hot (bandwidth-dominant)
  int C4 = C / 4;
  dim3 g2((C4 + 255) / 256, B);
  extract_labels_kernel<<<g2, 256, 0, stream>>>(onehot, labels, counts, C4);

  // 3) new_centers = centers
  size_t n4 = (size_t)C * D / 4;
  copy_centers_kernel<<<(unsigned)((n4 + 255) / 256), 256, 0, stream>>>(
      centers, new_centers, n4);

  // 4) WMMA Gram-diagonal squared distances + atomic centers scatter
  per_sample_kernel<<<B / 16, 32, 0, stream>>>(x, centers, labels, counts,
                                               result, new_centers);
}